// Binary_80788334837979
// MI455X (gfx1250) — compile-verified
//
#include <hip/hip_runtime.h>
#include <stdint.h>

// Greedy matching: one block per batch element b.
//   repeat N times: (r,c) = argmax over active rows/cols of s[b]; retire r,c; perm[r,c]=1
// LDS state: per-row running max over active columns + its argmax column.
// Phase 1: initial row maxima, streaming the matrix through LDS with
//          double-buffered TENSOR_LOAD_TO_LDS (TDM) transfers.
// Phase 2: 512 dependent iterations, 2 barriers each: wave32 two-level argmax
//          (second level replicated per-wave -> no broadcast barrier), then
//          fused retire+refresh.

#define N  512
#define NT 512            // threads per block = 16 wave32 waves
#define TILE_ROWS 16      // rows per TDM transfer (16 rows * 2KB = 32KB)
#define NTILES (N / TILE_ROWS)
#define TILE_BYTES (TILE_ROWS * N * 4)

static_assert(NT / 32 == TILE_ROWS, "one wave per staged row");

#if defined(__has_builtin)
#  if __has_builtin(__builtin_amdgcn_tensor_load_to_lds) && \
      __has_builtin(__builtin_amdgcn_s_wait_tensorcnt)
#    define HAVE_TDM 1
#  endif
#endif
#ifndef HAVE_TDM
#  define HAVE_TDM 0
#endif

typedef unsigned int v4u  __attribute__((ext_vector_type(4)));
typedef int          v8i_ __attribute__((ext_vector_type(8)));
typedef int          v4i_ __attribute__((ext_vector_type(4)));

// Issue one TDM transfer (no wait): TILE_ROWS x N fp32 tile, global -> LDS.
// Executed by exactly one wave (EXEC ignored; TDM is a per-wave DMA op).
__device__ __forceinline__ void tdm_issue_tile(const float* gsrc, uint32_t lds_byte_off) {
#if HAVE_TDM
  uint64_t ga = (uint64_t)(uintptr_t)gsrc;
  v4u g0;
  g0.x = 1u;                                      // count = 1 (valid user descriptor)
  g0.y = lds_byte_off;                            // lds_addr (bytes)
  g0.z = (uint32_t)(ga & 0xFFFFFFFFu);            // global_addr[31:0]
  g0.w = (uint32_t)((ga >> 32) & 0x1FFFFFFu)      // global_addr[56:32]
       | (2u << 30);                              // type = 2 ("image")
  v8i_ g1;
  g1[0] = (int)(2u << 16);                        // data_size = 2 -> 4 bytes
  g1[1] = (int)((N & 0xFFFFu) << 16);             // tensor_dim0[15:0] @ bits 63:48
  g1[2] = (int)(((N >> 16) & 0xFFFFu)             // tensor_dim0[31:16]
       | ((N & 0xFFFFu) << 16));                  // tensor_dim1[15:0]
  g1[3] = (int)(((N >> 16) & 0xFFFFu)             // tensor_dim1[31:16]
       | ((uint32_t)N << 16));                    // tile_dim0 = N
  g1[4] = TILE_ROWS;                              // tile_dim1 (tile_dim2 = 0)
  g1[5] = N;                                      // tensor_dim0_stride[31:0]
  g1[6] = 0;                                      // stride[47:32] | dim1_stride lo (2D)
  g1[7] = 0;
  v4i_ z4 = {0, 0, 0, 0};
#if __has_include(<hip/amd_detail/amd_gfx1250_TDM.h>)
  v8i_ z8 = {0, 0, 0, 0, 0, 0, 0, 0};
  __builtin_amdgcn_tensor_load_to_lds(g0, g1, z4, z4, z8, 0);   // clang-23 arity
#else
  __builtin_amdgcn_tensor_load_to_lds(g0, g1, z4, z4, 0);       // ROCm 7.2 arity
#endif
#else
  (void)gsrc; (void)lds_byte_off;
#endif
}

// Wave `wid` scans staged local row `wid`; lane-consecutive columns (bank-clean).
__device__ __forceinline__ void scan_tile_row(const float* tileBase, int t0,
                                              int lane, int wid,
                                              float* rowMax, int* rowArg) {
  const float NEG = -__builtin_inff();
  const float* row = tileBase + wid * N;
  float m = NEG; int mi = 0;
  for (int j = lane; j < N; j += 32) {
    float v = row[j];
    if (v > m) { m = v; mi = j; }               // strict > keeps first occurrence
  }
  for (int off = 16; off > 0; off >>= 1) {      // wave32 argmax, tie -> smaller col
    float om = __shfl_xor(m, off, 32);
    int   oi = __shfl_xor(mi, off, 32);
    if (om > m || (om == m && oi < mi)) { m = om; mi = oi; }
  }
  if (lane == 0) { rowMax[t0 + wid] = m; rowArg[t0 + wid] = mi; }
}

__global__ __launch_bounds__(NT)
void greedy_match_kernel(const float* __restrict__ s, float* __restrict__ out) {
  extern __shared__ char smem_raw[];
  float*    buf0    = (float*)smem_raw;                         // TILE_BYTES
  float*    buf1    = (float*)(smem_raw + TILE_BYTES);          // TILE_BYTES
  float*    rowMax  = (float*)(smem_raw + 2 * TILE_BYTES);      // N f32
  int*      rowArg  = (int*)  (smem_raw + 2 * TILE_BYTES + N * 4);
  unsigned* colMask = (unsigned*)(smem_raw + 2 * TILE_BYTES + N * 8);
  float*    redV    = (float*)((char*)colMask + (N / 32) * 4);  // 16 f32
  int*      redI    = (int*)  ((char*)redV + 16 * 4);           // 16 i32

  const int tid  = threadIdx.x;
  const int lane = tid & 31;
  const int wid  = tid >> 5;
  const int b    = blockIdx.x;
  const float* __restrict__ sb = s   + (size_t)b * N * N;
  float*       __restrict__ ob = out + (size_t)b * N * N;
  const float NEG = -__builtin_inff();

  if (tid < N / 32) colMask[tid] = 0xFFFFFFFFu;
  __syncthreads();

  // ---- Phase 1: initial per-row maxima (double-buffered TDM streaming) ----
#if HAVE_TDM
  if (wid == 0) tdm_issue_tile(sb, 0u);                         // tile 0 -> buf0
  for (int t = 0; t < NTILES; ++t) {
    if (wid == 0) {
      if (t + 1 < NTILES) {
        tdm_issue_tile(sb + (size_t)(t + 1) * TILE_ROWS * N,
                       (uint32_t)(((t + 1) & 1) * TILE_BYTES)); // prefetch next
        __builtin_amdgcn_s_wait_tensorcnt(1);                   // tile t complete
      } else {
        __builtin_amdgcn_s_wait_tensorcnt(0);                   // last tile complete
      }
    }
    __syncthreads();                        // DMA of tile t visible to all waves
    scan_tile_row((t & 1) ? buf1 : buf0, t * TILE_ROWS, lane, wid, rowMax, rowArg);
    __syncthreads();                        // buffer safe to overwrite next issue
  }
#else
  for (int t = 0; t < NTILES; ++t) {
    for (int i = tid; i < TILE_ROWS * N / 4; i += NT)
      ((float4*)buf0)[i] = ((const float4*)(sb + (size_t)t * TILE_ROWS * N))[i];
    __syncthreads();
    scan_tile_row(buf0, t * TILE_ROWS, lane, wid, rowMax, rowArg);
    __syncthreads();
  }
#endif

  // ---- Phase 2: N dependent greedy iterations, 2 barriers each ----
  for (int it = 0; it < N; ++it) {
    // level 1: each wave argmax-reduces its 32 row maxima (tie -> smaller row)
    float m = rowMax[tid];
    int   r = tid;
    for (int off = 16; off > 0; off >>= 1) {
      float om = __shfl_xor(m, off, 32);
      int   ri = __shfl_xor(r, off, 32);
      if (om > m || (om == m && ri < r)) { m = om; r = ri; }
    }
    if (lane == 0) { redV[wid] = m; redI[wid] = r; }
    __syncthreads();

    // level 2: EVERY wave reduces the 16 partials identically -> uniform rsel,
    // no broadcast barrier needed (deterministic redundant computation).
    float m2 = (lane < NT / 32) ? redV[lane] : NEG;
    int   r2 = (lane < NT / 32) ? redI[lane] : 0x7FFFFFFF;
    for (int off = 16; off > 0; off >>= 1) {
      float om = __shfl_xor(m2, off, 32);
      int   ri = __shfl_xor(r2, off, 32);
      if (om > m2 || (om == m2 && ri < r2)) { m2 = om; r2 = ri; }
    }
    const int rsel = r2;                 // identical across all lanes and waves
    const int csel = rowArg[rsel];

    // emit the selected row one-hot (coalesced; fully covers poisoned d_out)
    ob[(size_t)rsel * N + tid] = (tid == csel) ? 1.0f : 0.0f;

    // fused retire + refresh. Thread 0's mask store races only with readers of
    // the same 32-bit word, which locally clear csel anyway -> both orders OK.
    if (tid == 0) colMask[csel >> 5] &= ~(1u << (csel & 31));
    if (tid == rsel) {
      rowMax[tid] = NEG;                 // retire selected row
    } else if (rowMax[tid] != NEG && rowArg[tid] == csel) {
      const float* row = sb + (size_t)tid * N;   // 2KB row, hot in L2
      float mm = NEG; int mj = 0;
      for (int w = 0; w < N / 32; ++w) {
        unsigned msk = colMask[w];
        if (w == (csel >> 5)) msk &= ~(1u << (csel & 31));  // local retire of csel
        while (msk) {                    // ascending bits -> first-index ties
          int bit = __builtin_ctz(msk);
          msk &= msk - 1;
          float v = row[(w << 5) + bit];
          if (v > mm) { mm = v; mj = (w << 5) + bit; }
        }
      }
      rowMax[tid] = mm; rowArg[tid] = mj;
    }
    __syncthreads();
  }
}

extern "C" void kernel_launch(void* const* d_in, const int* in_sizes, int n_in,
                              void* d_out, int out_size, void* d_ws, size_t ws_size,
                              hipStream_t stream) {
  (void)n_in; (void)out_size; (void)d_ws; (void)ws_size;
  const float* s = (const float*)d_in[0];
  float* out = (float*)d_out;
  const int B = in_sizes[0] / (N * N);   // 256
  const size_t smem = (size_t)2 * TILE_BYTES      // double-buffered staging tiles
                    + (size_t)N * 4               // rowMax
                    + (size_t)N * 4               // rowArg
                    + (size_t)(N / 32) * 4        // colMask
                    + 16 * 4 + 16 * 4;            // reduction scratch
  greedy_match_kernel<<<B, NT, smem, stream>>>(s, out);
}